// OTSolver_15358803050559
// MI455X (gfx1250) — compile-verified
//
#include <hip/hip_runtime.h>
#include <hip/hip_bf16.h>
#include <math.h>

typedef __attribute__((ext_vector_type(2))) float v2f;
typedef __attribute__((ext_vector_type(8))) float v8f;

#define LQ 1024
#define BQ 32
#define DQ 256
#define LAMQ 0.2f

// ---------------------------------------------------------------------------
// Kernel 0: inverse row norms of m1 / m2.  m{1,2} layout (L,B,D); flat row id
// = i*B + b.  invn indexed [b*L + i].  One wave (32 lanes) per row, 8 floats
// per lane (two float4 loads), full-wave shfl reduction.
// ---------------------------------------------------------------------------
__global__ void k_norms(const float* __restrict__ m1, const float* __restrict__ m2,
                        float* __restrict__ invnx, float* __restrict__ invny) {
    int wave = blockIdx.x * (blockDim.x >> 5) + (threadIdx.x >> 5);
    int lane = threadIdx.x & 31;
    int which = wave >> 15;            // 0 -> m1, 1 -> m2 (32768 rows each)
    int rowid = wave & 32767;          // rowid = i*32 + b
    int i = rowid >> 5;
    int b = rowid & 31;
    const float* src = (which == 0) ? m1 : m2;
    const float* row = src + (size_t)rowid * DQ;
    float4 a = *(const float4*)(row + lane * 8);
    float4 c = *(const float4*)(row + lane * 8 + 4);
    float ss = a.x*a.x + a.y*a.y + a.z*a.z + a.w*a.w
             + c.x*c.x + c.y*c.y + c.z*c.z + c.w*c.w;
    for (int off = 16; off; off >>= 1) ss += __shfl_xor(ss, off, 32);
    if (lane == 0) {
        float inv = 1.0f / fmaxf(sqrtf(ss), 1e-5f);
        (which ? invny : invnx)[b * LQ + i] = inv;
    }
}

// ---------------------------------------------------------------------------
// Epilogue for one 16x16 WMMA tile: M = 1 - dot (+100 off-band), E = exp(-lam*M)
// masked to valid block.  C/D layout: lane l, VGPR v -> row = v + 8*(l>>4),
// col = l&15.
// ---------------------------------------------------------------------------
__device__ __forceinline__ void tile_epilogue(const v8f& acc, int rowt, int colt,
                                              int l, int ml, int h,
                                              float* __restrict__ Eb) {
    const int j = colt + ml;
    #pragma unroll
    for (int v = 0; v < 8; ++v) {
        int i = rowt + v + 8 * h;
        float Mv = 1.0f - acc[v];
        int d = i - j; if (d < 0) d = -d;
        if (d > 2) Mv += 100.0f;
        float e = ((i < l) && (j < l)) ? __expf(-LAMQ * Mv) : 0.0f;
        Eb[(size_t)i * LQ + j] = e;
    }
}

// ---------------------------------------------------------------------------
// Kernel 1: E[b,i,j] via fp32 WMMA 16x16x4, 2x2 macro-tile (32x32) per wave.
// 4 independent accumulator chains -> 4 WMMAs per 4 fragment loads per K-step,
// halves A/B L2 traffic vs 1 tile/wave and hides WMMA RAW latency.
// A-frag (16x4 f32): lane l: m=l&15; VGPR0/1 hold K = k0+2*(l>>4) + {0,1}.
// B-frag (4x16 f32): lane l: n=l&15; VGPR0/1 hold K = k0+2*(l>>4) + {0,1}.
// ---------------------------------------------------------------------------
__global__ void k_gemm_e(const float* __restrict__ m1, const float* __restrict__ m2,
                         const int* __restrict__ lengths,
                         const float* __restrict__ invnx, const float* __restrict__ invny,
                         float* __restrict__ E) {
    const int lane = threadIdx.x & 31;
    const int wave = blockIdx.x * 8 + (threadIdx.x >> 5);  // blockDim == 256
    const int b    = wave >> 10;                           // 1024 macro-tiles/batch
    const int t    = wave & 1023;                          // 32 x 32 macro-tiles
    const int row0 = (t >> 5) << 5;
    const int col0 = (t & 31) << 5;
    const int l    = lengths[b];
    const int ml   = lane & 15;
    const int h    = lane >> 4;

    const int gi0 = row0 + ml,      gi1 = row0 + 16 + ml;
    const int gj0 = col0 + ml,      gj1 = col0 + 16 + ml;
    const float inx0 = invnx[b * LQ + gi0], inx1 = invnx[b * LQ + gi1];
    const float iny0 = invny[b * LQ + gj0], iny1 = invny[b * LQ + gj1];
    const float* ap0 = m1 + (size_t)gi0 * (BQ * DQ) + b * DQ + 2 * h;
    const float* ap1 = m1 + (size_t)gi1 * (BQ * DQ) + b * DQ + 2 * h;
    const float* bp0 = m2 + (size_t)gj0 * (BQ * DQ) + b * DQ + 2 * h;
    const float* bp1 = m2 + (size_t)gj1 * (BQ * DQ) + b * DQ + 2 * h;

    v8f a00 = {}, a01 = {}, a10 = {}, a11 = {};
    #pragma unroll 4
    for (int k0 = 0; k0 < DQ; k0 += 4) {
        v2f fa0 = *(const v2f*)(ap0 + k0) * inx0;
        v2f fa1 = *(const v2f*)(ap1 + k0) * inx1;
        v2f fb0 = *(const v2f*)(bp0 + k0) * iny0;
        v2f fb1 = *(const v2f*)(bp1 + k0) * iny1;
        a00 = __builtin_amdgcn_wmma_f32_16x16x4_f32(false, fa0, false, fb0, (short)0, a00, false, false);
        a01 = __builtin_amdgcn_wmma_f32_16x16x4_f32(false, fa0, false, fb1, (short)0, a01, false, false);
        a10 = __builtin_amdgcn_wmma_f32_16x16x4_f32(false, fa1, false, fb0, (short)0, a10, false, false);
        a11 = __builtin_amdgcn_wmma_f32_16x16x4_f32(false, fa1, false, fb1, (short)0, a11, false, false);
    }

    float* Eb = E + (size_t)b * LQ * LQ;
    tile_epilogue(a00, row0,      col0,      l, ml, h, Eb);
    tile_epilogue(a01, row0,      col0 + 16, l, ml, h, Eb);
    tile_epilogue(a10, row0 + 16, col0,      l, ml, h, Eb);
    tile_epilogue(a11, row0 + 16, col0 + 16, l, ml, h, Eb);
}

// ---------------------------------------------------------------------------
// Kernel 2: deterministic row reductions of E.
//   R[b,i]  = sum_j E[b,i,j]
//   CM[b,i] = sum_j E*M     with M recovered exactly as -ln(E)/lam = -5 ln E
// One wave per row (coalesced float4 reads).
// ---------------------------------------------------------------------------
__global__ void k_rowsum(const float* __restrict__ E,
                         float* __restrict__ R, float* __restrict__ CM) {
    int wave = blockIdx.x * 8 + (threadIdx.x >> 5);   // wave = b*1024 + i
    int lane = threadIdx.x & 31;
    const float* row = E + (size_t)wave * LQ;
    float se = 0.0f, sm = 0.0f;
    for (int j = lane * 4; j < LQ; j += 128) {
        float4 e4 = *(const float4*)(row + j);
        float ev[4] = {e4.x, e4.y, e4.z, e4.w};
        #pragma unroll
        for (int q = 0; q < 4; ++q) {
            float e = ev[q];
            se += e;
            if (e > 0.0f) sm += e * (-5.0f * __logf(e));
        }
    }
    for (int off = 16; off; off >>= 1) {
        se += __shfl_xor(se, off, 32);
        sm += __shfl_xor(sm, off, 32);
    }
    if (lane == 0) { R[wave] = se; CM[wave] = sm; }
}

// ---------------------------------------------------------------------------
// Kernel 3: collapsed Sinkhorn. Because BOTH scale factors in the reference
// are applied along rows, P == diag(f)*E throughout. State per row: f_i, u_i.
// Per iteration: usum=f*R; EPS test (block max); f1 = mask ? f*(1/l)/usum : 0;
// s_j = sum_i f1_i E[i,j] (coalesced matvec, f1 broadcast from LDS);
// f = mask ? f1*(1/l)/s_i : 0.  One block (1024 threads) per batch.
// ---------------------------------------------------------------------------
__global__ void __launch_bounds__(1024)
k_sinkhorn(const float* __restrict__ E, const int* __restrict__ lengths,
           const float* __restrict__ R, const float* __restrict__ CM,
           float* __restrict__ fv, float* __restrict__ invrs,
           float* __restrict__ cost) {
    __shared__ float sh[LQ];
    __shared__ float red[32];
    const int b = blockIdx.x;
    const int i = threadIdx.x;
    const int lane = i & 31, wid = i >> 5;
    const int l = lengths[b];
    const bool maski = i < l;
    const float Ri = R[b * LQ + i];
    const float Ci = CM[b * LQ + i];
    const float* Eb = E + (size_t)b * LQ * LQ;

    // S = sum(R) -> f0 = 1/S  (P0 = E/S)
    float v = Ri;
    for (int off = 16; off; off >>= 1) v += __shfl_xor(v, off, 32);
    if (lane == 0) red[wid] = v;
    __syncthreads();
    if (wid == 0) {
        float r = red[lane];
        for (int off = 16; off; off >>= 1) r += __shfl_xor(r, off, 32);
        if (lane == 0) red[0] = r;
    }
    __syncthreads();
    const float S = red[0];
    __syncthreads();

    float f = 1.0f / S;
    float u = 0.0f;
    const float rmarg = 1.0f / (float)l;

    for (int it = 0; it < 10; ++it) {
        float usum = f * Ri;
        float m = fabsf(u - usum);
        for (int off = 16; off; off >>= 1) m = fmaxf(m, __shfl_xor(m, off, 32));
        if (lane == 0) red[wid] = m;
        __syncthreads();
        if (wid == 0) {
            float r = red[lane];
            for (int off = 16; off; off >>= 1) r = fmaxf(r, __shfl_xor(r, off, 32));
            if (lane == 0) red[0] = r;
        }
        __syncthreads();
        float mx = red[0];
        __syncthreads();
        if (mx <= 1e-7f) break;                      // block-uniform

        float f1 = maski ? f * (rmarg / usum) : 0.0f;
        sh[i] = f1;
        __syncthreads();
        float s = 0.0f;
        for (int k = 0; k < l; ++k)                  // f1[k>=l]==0, E rows 0
            s += sh[k] * Eb[(size_t)k * LQ + i];     // coalesced across i
        f = maski ? f1 * (rmarg / s) : 0.0f;
        u = usum;
        __syncthreads();
    }

    fv[b * LQ + i] = f;
    invrs[b * LQ + i] = maski ? 1.0f / (f * Ri) : 1.0f;  // rs=where(mask,f*R,1)

    // cost = sum_i f_i * CM_i
    float cv = f * Ci;
    for (int off = 16; off; off >>= 1) cv += __shfl_xor(cv, off, 32);
    if (lane == 0) red[wid] = cv;
    __syncthreads();
    if (wid == 0) {
        float r = red[lane];
        for (int off = 16; off; off >>= 1) r += __shfl_xor(r, off, 32);
        if (lane == 0) cost[b] = r;
    }
}

// ---------------------------------------------------------------------------
// Kernel 4: in-place finalize  P[b,i,j] = mm ? f_i * E[b,i,j] * invrs_j : 0
// (reference divides column j by rowsum[j]).  float4 per thread.
// ---------------------------------------------------------------------------
__global__ void k_finalize(float* __restrict__ P, const int* __restrict__ lengths,
                           const float* __restrict__ fv, const float* __restrict__ invrs) {
    size_t idx = ((size_t)blockIdx.x * blockDim.x + threadIdx.x) * 4;
    int b = (int)(idx >> 20);                  // L*L = 2^20
    size_t rem = idx & (size_t)1048575;
    int i = (int)(rem >> 10);
    int j = (int)(rem & 1023);
    int l = lengths[b];
    float fi = fv[b * LQ + i];
    const float* ir = invrs + b * LQ + j;
    float4 e = *(const float4*)(P + idx);
    bool vi = i < l;
    float4 o;
    o.x = (vi && (j + 0) < l) ? fi * e.x * ir[0] : 0.0f;
    o.y = (vi && (j + 1) < l) ? fi * e.y * ir[1] : 0.0f;
    o.z = (vi && (j + 2) < l) ? fi * e.z * ir[2] : 0.0f;
    o.w = (vi && (j + 3) < l) ? fi * e.w * ir[3] : 0.0f;
    *(float4*)(P + idx) = o;
}

// ---------------------------------------------------------------------------
extern "C" void kernel_launch(void* const* d_in, const int* in_sizes, int n_in,
                              void* d_out, int out_size, void* d_ws, size_t ws_size,
                              hipStream_t stream) {
    const float* m1      = (const float*)d_in[0];
    const float* m2      = (const float*)d_in[1];
    const int*   lengths = (const int*)d_in[2];

    float* P    = (float*)d_out;                 // (B,L,L) — used as E scratch first
    float* cost = P + (size_t)BQ * LQ * LQ;      // (B,)

    // small workspace: 6 * 32768 floats = 768 KB
    float* wf    = (float*)d_ws;
    float* invnx = wf;
    float* invny = wf + 32768;
    float* R     = wf + 65536;
    float* CM    = wf + 98304;
    float* fvv   = wf + 131072;
    float* invrs = wf + 163840;

    // 2*B*L = 65536 rows, 8 waves/block
    k_norms   <<<8192, 256, 0, stream>>>(m1, m2, invnx, invny);
    // 32 batches * 1024 macro-tiles = 32768 waves, 8 waves/block
    k_gemm_e  <<<4096, 256, 0, stream>>>(m1, m2, lengths, invnx, invny, P);
    // B*L = 32768 rows, 8 waves/block
    k_rowsum  <<<4096, 256, 0, stream>>>(P, R, CM);
    // one block per batch
    k_sinkhorn<<<BQ, 1024, 0, stream>>>(P, lengths, R, CM, fvv, invrs, cost);
    // B*L*L / 4 elems per thread / 256 threads
    k_finalize<<<32768, 256, 0, stream>>>(P, lengths, fvv, invrs);
}